// LossFun_44186623541345
// MI455X (gfx1250) — compile-verified
//
#include <hip/hip_runtime.h>
#include <hip/hip_bf16.h>

typedef __attribute__((ext_vector_type(4))) float        v4f;
typedef __attribute__((ext_vector_type(4))) unsigned int u32x4;
typedef __attribute__((ext_vector_type(8))) int          i32x8;
typedef __attribute__((ext_vector_type(4))) int          i32x4;

#define VOCAB   32000
#define NROWS   (8 * 2048)   /* B * L */
#define INV_B   0.125f       /* 1/B, B=8 */

#if __has_builtin(__builtin_amdgcn_tensor_load_to_lds) && __has_builtin(__builtin_amdgcn_s_wait_tensorcnt)
#define USE_TDM 1
#else
#define USE_TDM 0
#endif

// ---------------------------------------------------------------------------
// Pass 1: one wave32 per row. Streaming argmax over V=32000 fp32 scores with
// non-temporal b128 loads, then wave shuffle reduction. Lane 0 writes the
// per-row loss contribution to the workspace.
// ---------------------------------------------------------------------------
__global__ __launch_bounds__(256) void rowArgmaxKernel(
    const float* __restrict__ scores,
    const int*   __restrict__ targets,
    float*       __restrict__ ws)
{
    const int lane = threadIdx.x & 31;
    const int row  = blockIdx.x * 8 + (threadIdx.x >> 5);   // 8 waves / block

    const v4f* __restrict__ rp = (const v4f*)(scores + (size_t)row * VOCAB);

    // Four independent (max, idx) chains -> 4x shorter dependency chains.
    float b0 = -1.0f, b1 = -1.0f, b2 = -1.0f, b3 = -1.0f;  // scores in (0,1)
    int   i0 = 0, i1 = 1, i2 = 2, i3 = 3;

    #pragma unroll 5
    for (int i = lane; i < VOCAB / 4; i += 32) {
        v4f v = __builtin_nontemporal_load(&rp[i]);         // TH=NT: single-use stream
        const int base = i << 2;
        if (v.x > b0) { b0 = v.x; i0 = base;     }
        if (v.y > b1) { b1 = v.y; i1 = base + 1; }
        if (v.z > b2) { b2 = v.z; i2 = base + 2; }
        if (v.w > b3) { b3 = v.w; i3 = base + 3; }
    }

    // Merge the 4 chains; ties -> smallest index (jnp.argmax semantics).
    float best = b0; int bidx = i0;
    if (b1 > best || (b1 == best && i1 < bidx)) { best = b1; bidx = i1; }
    if (b2 > best || (b2 == best && i2 < bidx)) { best = b2; bidx = i2; }
    if (b3 > best || (b3 == best && i3 < bidx)) { best = b3; bidx = i3; }

    // Wave32 butterfly reduction (ds_swizzle/permute path; no LDS memory).
    #pragma unroll
    for (int off = 16; off > 0; off >>= 1) {
        float ov = __shfl_xor(best, off, 32);
        int   oi = __shfl_xor(bidx, off, 32);
        if (ov > best || (ov == best && oi < bidx)) { best = ov; bidx = oi; }
    }

    if (lane == 0) {
        const int   t = targets[row];
        const float w = (t == 0) ? 1.0f : 2.0f;             // BETA = 2
        ws[row] = (bidx == t) ? (w * logf(best)) : 0.0f;
    }
}

// ---------------------------------------------------------------------------
// Pass 2: deterministic finalize. One block; the 64 KB workspace is DMA'd
// into LDS by the Tensor Data Mover (one TENSOR_LOAD_TO_LDS descriptor),
// then reduced in a fixed order. out[0] = -sum / B.
// ---------------------------------------------------------------------------
__global__ __launch_bounds__(256) void finalizeKernel(
    const float* __restrict__ ws,
    float*       __restrict__ out)
{
    __shared__ float smem[NROWS];                           // 64 KB of 320 KB LDS
    const int chunk = NROWS / 256;                          // 64 values / thread
    const int basei = threadIdx.x * chunk;
    float s = 0.0f;

#if USE_TDM
    if (threadIdx.x < 32) {                                 // one wave issues the DMA
        // Generic LDS pointer carries the LDS byte offset in addr[31:0].
        const unsigned lds_addr = (unsigned)(unsigned long long)(&smem[0]);
        const unsigned long long ga = (unsigned long long)ws;

        // D# group 0: count=1 (valid user descriptor), lds_addr,
        // 57-bit global addr, type=2 ("image") in bits [127:126].
        u32x4 g0;
        g0[0] = 1u;
        g0[1] = lds_addr;
        g0[2] = (unsigned)(ga & 0xFFFFFFFFu);
        g0[3] = (unsigned)((ga >> 32) & 0x01FFFFFFu) | (2u << 30);

        // D# group 1: wg_mask=0 (not in cluster), data_size=2 (4 bytes),
        // 1-D tile: tensor_dim0 = tile_dim0 = NROWS, tile_dim1 = 0 (unused).
        i32x8 g1;
        g1[0] = (2 << 16);                                  // data_size = 4B
        g1[1] = (int)((NROWS & 0xFFFF) << 16);              // tensor_dim0[15:0]
        g1[2] = (int)(((NROWS >> 16) & 0xFFFF) | (1 << 16));// dim0[31:16], tensor_dim1=1
        g1[3] = (int)((NROWS & 0xFFFF) << 16);              // tile_dim0 = NROWS
        g1[4] = 0;                                          // tile_dim1/2 unused
        g1[5] = NROWS;                                      // tensor_dim0_stride lo
        g1[6] = 0;
        g1[7] = 0;

        i32x4 gz4 = {0, 0, 0, 0};                           // groups 2/3 unused (<=2D)
        i32x8 gz8 = {0, 0, 0, 0, 0, 0, 0, 0};               // extra group (clang-23 form)
        __builtin_amdgcn_tensor_load_to_lds(g0, g1, gz4, gz4, gz8, 0);
        __builtin_amdgcn_s_wait_tensorcnt(0);               // TENSORcnt == 0
    }
    __syncthreads();

    #pragma unroll 8
    for (int i = 0; i < chunk; ++i) s += smem[basei + i];
    __syncthreads();                                        // done reading staged data
#else
    #pragma unroll 8
    for (int i = 0; i < chunk; ++i) s += ws[basei + i];
#endif

    // Fixed-order tree reduction -> deterministic result.
    smem[threadIdx.x] = s;
    __syncthreads();
    #pragma unroll
    for (int st = 128; st > 0; st >>= 1) {
        if ((int)threadIdx.x < st) smem[threadIdx.x] += smem[threadIdx.x + st];
        __syncthreads();
    }
    if (threadIdx.x == 0) out[0] = -smem[0] * INV_B;
}

// ---------------------------------------------------------------------------
extern "C" void kernel_launch(void* const* d_in, const int* in_sizes, int n_in,
                              void* d_out, int out_size, void* d_ws, size_t ws_size,
                              hipStream_t stream)
{
    const float* scores  = (const float*)d_in[0];   // [8, 2048, 32000] fp32
    const int*   targets = (const int*)d_in[1];     // [8, 2048] int
    float*       ws      = (float*)d_ws;            // NROWS per-row contributions
    float*       out     = (float*)d_out;           // shape (1,)

    rowArgmaxKernel<<<NROWS / 8, 256, 0, stream>>>(scores, targets, ws);
    finalizeKernel<<<1, 256, 0, stream>>>(ws, out);
}